// LNODEFunc9_83159156785560
// MI455X (gfx1250) — compile-verified
//
#include <hip/hip_runtime.h>
#include <math.h>

typedef __attribute__((ext_vector_type(2))) float v2f;
typedef __attribute__((ext_vector_type(8))) float v8f;

#define LN_EPS 1e-6f

// ---------- prep: fold all F=20 feature math into 31 constants ----------
__device__ __forceinline__ void killing_branch(const float* __restrict__ wlin,  // [3,20]
                                               const float* __restrict__ wdir,  // [3,3]
                                               const float Rm[20][3],
                                               float M[3][3]) {
  float lin[3][3];
  for (int g = 0; g < 3; ++g)
    for (int k = 0; k < 3; ++k) {
      float acc = 0.f;
      for (int f = 0; f < 20; ++f) acc += wlin[g * 20 + f] * Rm[f][k];
      lin[g][k] = acc;
    }
  float d[3][3];
  for (int g = 0; g < 3; ++g)
    for (int k = 0; k < 3; ++k)
      d[g][k] = wdir[g * 3 + 0] * lin[0][k] + wdir[g * 3 + 1] * lin[1][k] +
                wdir[g * 3 + 2] * lin[2][k];
  float feat[3][3];
  for (int g = 0; g < 3; ++g) {
    float kf = -2.f * (lin[g][0] * d[g][0] + lin[g][1] * d[g][1] + lin[g][2] * d[g][2]);
    float dd = -2.f * (d[g][0] * d[g][0] + d[g][1] * d[g][1] + d[g][2] * d[g][2]);
    float s = kf / (dd + LN_EPS);
    for (int k = 0; k < 3; ++k)
      feat[g][k] = (kf < 0.f) ? lin[g][k] : (lin[g][k] - s * d[g][k]);
  }
  for (int k = 0; k < 3; ++k)
    for (int l = 0; l < 3; ++l)
      M[k][l] = feat[0][k] * feat[0][l] + feat[1][k] * feat[1][l] + feat[2][k] * feat[2][l];
}

// ws layout (floats): [0..8]=M1, [9..17]=M2, [18]=A, [19..27]=L, [28..30]=c
__global__ void lnode_prep_kernel(const float* __restrict__ t, const float* __restrict__ R,
                                  const float* __restrict__ m, const float* __restrict__ w1_lin,
                                  const float* __restrict__ w1_dir, const float* __restrict__ w2_lin,
                                  const float* __restrict__ w2_dir, const float* __restrict__ wb,
                                  const float* __restrict__ wo, float* __restrict__ ws) {
  if (threadIdx.x != 0 || blockIdx.x != 0) return;
  float tv = t[0];
  float sv = sinf(tv), cv = cosf(tv);
  float ui0 = 10.f * sv, ui1 = 10.f * cv, ui2 = 10.f * sv;
  float u[3];
  for (int k = 0; k < 3; ++k)
    u[k] = R[k * 3 + 0] * ui0 + R[k * 3 + 1] * ui1 + R[k * 3 + 2] * ui2;

  float Rm[20][3];  // (R @ m)^T : feature f, component k
  for (int f = 0; f < 20; ++f)
    for (int k = 0; k < 3; ++k)
      Rm[f][k] = R[k * 3 + 0] * m[0 * 20 + f] + R[k * 3 + 1] * m[1 * 20 + f] +
                 R[k * 3 + 2] * m[2 * 20 + f];

  float M1[3][3], M2[3][3];
  killing_branch(w1_lin, w1_dir, Rm, M1);
  killing_branch(w2_lin, w2_dir, Rm, M2);

  float As = 0.f, Bs = 0.f, Cs = 0.f, Ds = 0.f, Es = 0.f;
  for (int g = 0; g < 20; ++g) {
    float a = wb[g * 2 + 0], b = wb[g * 2 + 1], w = wo[g];
    As += w * a * a; Bs += w * a * b; Cs += w * b * b; Ds += w * a; Es += w * b;
  }
  float p[3], q[3];
  for (int k = 0; k < 3; ++k) {
    p[k] = M1[k][0] * u[0] + M1[k][1] * u[1] + M1[k][2] * u[2];
    q[k] = M2[k][0] * u[0] + M2[k][1] * u[1] + M2[k][2] * u[2];
  }
  float hq[3][3] = {{0.f, -q[2], q[1]}, {q[2], 0.f, -q[0]}, {-q[1], q[0], 0.f}};
  float hp[3][3] = {{0.f, -p[2], p[1]}, {p[2], 0.f, -p[0]}, {-p[1], p[0], 0.f}};
  float L[3][3];
  for (int k = 0; k < 3; ++k)
    for (int l = 0; l < 3; ++l) {
      float acc = 0.f;
      for (int j = 0; j < 3; ++j) acc += -hq[k][j] * M1[j][l] + hp[k][j] * M2[j][l];
      L[k][l] = Bs * acc + ((k == l) ? Ds : 0.f);
    }
  float pq0 = p[1] * q[2] - p[2] * q[1];
  float pq1 = p[2] * q[0] - p[0] * q[2];
  float pq2 = p[0] * q[1] - p[1] * q[0];
  float c0 = Cs * pq0 + Es * u[0], c1 = Cs * pq1 + Es * u[1], c2 = Cs * pq2 + Es * u[2];

  for (int k = 0; k < 3; ++k)
    for (int l = 0; l < 3; ++l) {
      ws[k * 3 + l] = M1[k][l];
      ws[9 + k * 3 + l] = M2[k][l];
      ws[19 + k * 3 + l] = L[k][l];
    }
  ws[18] = As;
  ws[28] = c0; ws[29] = c1; ws[30] = c2;
}

// ---------- main streaming kernel: out = A*(M1x x M2x) + L x + c ----------
__global__ __launch_bounds__(256) void lnode_main_kernel(const float* __restrict__ x,
                                                         const float* __restrict__ ws,
                                                         float* __restrict__ out, int npts) {
  const int tid = blockIdx.x * blockDim.x + threadIdx.x;
  const int lane = threadIdx.x & 31;
  const bool lower = lane < 16;

  // A operand (16x4 f32, ISA 7.12.2): lanes 0-15 hold M=lane K={0,1};
  // lanes 16-31 hold M=lane-16 K={2,3}. W=[M1;M2] in rows 0-5 AND 8-13 so both
  // lane halves of D carry the 6 useful rows.
  const int Mrow = lane & 15;
  const int Mr = (Mrow < 6) ? Mrow : ((Mrow >= 8 && Mrow < 14) ? (Mrow - 8) : -1);
  const int K0 = lower ? 0 : 2;  // always < 3
  const bool k1_valid = lower;   // K1 = 1 valid, K1 = 3 is padding
  v2f va;
  va.x = (Mr >= 0) ? ws[Mr * 3 + K0] : 0.f;
  va.y = (Mr >= 0 && k1_valid) ? ws[Mr * 3 + 1] : 0.f;

  const float As = ws[18];
  const float L00 = ws[19], L01 = ws[20], L02 = ws[21];
  const float L10 = ws[22], L11 = ws[23], L12 = ws[24];
  const float L20 = ws[25], L21 = ws[26], L22 = ws[27];
  const float c0 = ws[28], c1 = ws[29], c2 = ws[30];

  // Each lane owns point tid; clamp so every lane issues a valid (coalesced b96) load.
  const int pidx = (tid < npts) ? tid : (npts - 1);
  const float x0 = x[3 * pidx + 0];
  const float x1 = x[3 * pidx + 1];
  const float x2 = x[3 * pidx + 2];

  // B operand (4x16 f32): column = lane&15; lanes 0-15 hold K={0,1}, 16-31 K={2,3}.
  // WMMA#1 columns = points of lanes 0-15, WMMA#2 columns = points of lanes 16-31.
  const float ox0 = __shfl_xor(x0, 16, 32);
  const float ox1 = __shfl_xor(x1, 16, 32);
  const float ox2 = __shfl_xor(x2, 16, 32);
  v2f vb1, vb2;
  vb1.x = lower ? x0 : ox2;   // K=0 row (lower) / K=2 row = z of lane-16's point
  vb1.y = lower ? x1 : 0.f;   // K=1 row / K=3 pad
  vb2.x = lower ? ox0 : x2;   // K=0 row = x of lane+16's point / K=2 row = own z
  vb2.y = lower ? ox1 : 0.f;  // K=1 row = y of lane+16's point / K=3 pad

  const v8f cz = {};
  v8f d1 = __builtin_amdgcn_wmma_f32_16x16x4_f32(false, va, false, vb1, (short)0, cz, false, false);
  v8f d2 = __builtin_amdgcn_wmma_f32_16x16x4_f32(false, va, false, vb2, (short)0, cz, false, false);

  // D layout: lane n (n<16) vgpr r = row r, col n  -> its own point (from WMMA#1);
  // lane 16+n vgpr r = row 8+r, col n              -> its own point (from WMMA#2).
  const float v1x = lower ? d1[0] : d2[0];
  const float v1y = lower ? d1[1] : d2[1];
  const float v1z = lower ? d1[2] : d2[2];
  const float w2x = lower ? d1[3] : d2[3];
  const float w2y = lower ? d1[4] : d2[4];
  const float w2z = lower ? d1[5] : d2[5];

  const float crx = v1y * w2z - v1z * w2y;
  const float cry = v1z * w2x - v1x * w2z;
  const float crz = v1x * w2y - v1y * w2x;

  const float o0 = As * crx + (L00 * x0 + L01 * x1 + L02 * x2 + c0);
  const float o1 = As * cry + (L10 * x0 + L11 * x1 + L12 * x2 + c1);
  const float o2 = As * crz + (L20 * x0 + L21 * x1 + L22 * x2 + c2);

  if (tid < npts) {
    out[3 * tid + 0] = o0;
    out[3 * tid + 1] = o1;
    out[3 * tid + 2] = o2;
  }
}

extern "C" void kernel_launch(void* const* d_in, const int* in_sizes, int n_in,
                              void* d_out, int out_size, void* d_ws, size_t ws_size,
                              hipStream_t stream) {
  const float* x      = (const float*)d_in[0];
  const float* t      = (const float*)d_in[1];
  const float* R      = (const float*)d_in[2];
  const float* m      = (const float*)d_in[3];
  const float* w1_lin = (const float*)d_in[4];
  const float* w1_dir = (const float*)d_in[5];
  const float* w2_lin = (const float*)d_in[6];
  const float* w2_dir = (const float*)d_in[7];
  const float* wb     = (const float*)d_in[8];
  const float* wo     = (const float*)d_in[9];
  float* out = (float*)d_out;
  float* ws  = (float*)d_ws;

  const int npts = in_sizes[0] / 3;

  lnode_prep_kernel<<<1, 32, 0, stream>>>(t, R, m, w1_lin, w1_dir, w2_lin, w2_dir, wb, wo, ws);

  const int threads = 256;  // 8 waves per block on wave32
  const int blocks = (npts + threads - 1) / threads;
  lnode_main_kernel<<<blocks, threads, 0, stream>>>(x, ws, out, npts);
}